// Renderer_35210141892885
// MI455X (gfx1250) — compile-verified
//
#include <hip/hip_runtime.h>
#include <math.h>

#define S 129
#define NS3 (S * S * S)
#define LP 224
#define NL3 (LP * LP * LP)
#define CABS 0.2f

struct M3 { float m[9]; };

typedef __attribute__((ext_vector_type(2))) float v2f;
typedef __attribute__((ext_vector_type(8))) float v8f;

// ---------------- gather helpers (zero-padded, reference semantics) --------
__device__ __forceinline__ float sample_d(const float* __restrict__ vol, int zi, int yi, int xi) {
    if ((unsigned)zi < (unsigned)S && (unsigned)yi < (unsigned)S && (unsigned)xi < (unsigned)S)
        return vol[(zi * S + yi) * S + xi];
    return 0.0f;
}
__device__ __forceinline__ float sample_L(const float* __restrict__ vol, int zi, int yi, int xi) {
    if ((unsigned)zi < (unsigned)LP && (unsigned)yi < (unsigned)LP && (unsigned)xi < (unsigned)LP)
        return vol[(zi * LP + yi) * LP + xi];
    return 0.0f;
}

// ---------------- Kernel A: forward rotation of the padded density ---------
// rot[dd][hh][ww] = trilinear sample of zero-padded d at R1 * base(dd,hh,ww)
__global__ __launch_bounds__(256) void k_rotate_fwd(const float* __restrict__ d,
                                                    float* __restrict__ rot, M3 R) {
    int idx = blockIdx.x * 256 + threadIdx.x;       // exact grid: NL3 threads
    int ww = idx % LP;
    int t  = idx / LP;
    int hh = t % LP;
    int dd = t / LP;
    const float Gf   = 0.5f * (float)(LP - 1);      // 111.5
    const float invG = 1.0f / Gf;
    float bx = ww * invG - 1.0f;
    float by = hh * invG - 1.0f;
    float bz = dd * invG - 1.0f;
    // source coords directly in d-space (padded coord minus 48)
    float xs = (R.m[0] * bx + R.m[1] * by + R.m[2] * bz + 1.0f) * Gf - 48.0f;
    float ys = (R.m[3] * bx + R.m[4] * by + R.m[5] * bz + 1.0f) * Gf - 48.0f;
    float zs = (R.m[6] * bx + R.m[7] * by + R.m[8] * bz + 1.0f) * Gf - 48.0f;
    float v = 0.0f;
    if (xs > -1.0f && xs < (float)S &&
        ys > -1.0f && ys < (float)S &&
        zs > -1.0f && zs < (float)S) {
        float fx = floorf(xs), fy = floorf(ys), fz = floorf(zs);
        int x0 = (int)fx, y0 = (int)fy, z0 = (int)fz;
        float wx = xs - fx, wy = ys - fy, wz = zs - fz;
        float c000 = sample_d(d, z0,   y0,   x0  ), c001 = sample_d(d, z0,   y0,   x0+1);
        float c010 = sample_d(d, z0,   y0+1, x0  ), c011 = sample_d(d, z0,   y0+1, x0+1);
        float c100 = sample_d(d, z0+1, y0,   x0  ), c101 = sample_d(d, z0+1, y0,   x0+1);
        float c110 = sample_d(d, z0+1, y0+1, x0  ), c111 = sample_d(d, z0+1, y0+1, x0+1);
        float l00 = c000 + wx * (c001 - c000);
        float l01 = c010 + wx * (c011 - c010);
        float l10 = c100 + wx * (c101 - c100);
        float l11 = c110 + wx * (c111 - c110);
        float l0  = l00 + wy * (l01 - l00);
        float l1  = l10 + wy * (l11 - l10);
        v = l0 + wz * (l1 - l0);
    }
    rot[idx] = v;
}

// ---------------- Kernel B: WMMA suffix-scan + exp along depth -------------
// In-place: rot <- exp(-CABS * revcumsum_dd(rot)). One wave handles a
// (hh, 16-wide ww block) slab. Per 16-deep tile: Y = U * X via 4 chained
// V_WMMA_F32_16X16X4_F32, where U is upper-triangular ones (suffix-sum matrix).
__global__ __launch_bounds__(256) void k_scan_exp(float* __restrict__ rot) {
    int lane = threadIdx.x & 31;
    int gw   = (blockIdx.x * 256 + threadIdx.x) >> 5;   // global wave id, [0, 224*14)
    int hh   = gw / 14;
    int wb   = gw % 14;
    int col  = lane & 15;
    int half = lane >> 4;          // 0: lanes 0-15, 1: lanes 16-31
    int koff = half * 2;           // per ISA f32 A/B lane layout
    int row  = col;                // A-matrix row held by this lane

    // A = U (16x16 upper triangular incl. diagonal), sliced into 4 K=4 pieces.
    // A layout (f32 16x4): lanes 0-15 hold K=koff, koff+1 in VGPR0/1; lanes 16-31 K=2,3.
    v2f a0, a1, a2, a3;
    a0.x = (0  + koff >= row) ? 1.0f : 0.0f;  a0.y = (1  + koff >= row) ? 1.0f : 0.0f;
    a1.x = (4  + koff >= row) ? 1.0f : 0.0f;  a1.y = (5  + koff >= row) ? 1.0f : 0.0f;
    a2.x = (8  + koff >= row) ? 1.0f : 0.0f;  a2.y = (9  + koff >= row) ? 1.0f : 0.0f;
    a3.x = (12 + koff >= row) ? 1.0f : 0.0f;  a3.y = (13 + koff >= row) ? 1.0f : 0.0f;

    const int wc = wb * 16 + col;
    float carry = 0.0f;
    for (int b = 13; b >= 0; --b) {
        int dd0 = b * 16;
        // B-matrix 4x16 slices: VGPR0 lanes0-15 = row koff, lanes16-31 = row koff (+2 via koff)
        v2f b0, b1, b2, b3;
        b0.x = rot[((dd0 +  0 + koff) * LP + hh) * LP + wc];
        b0.y = rot[((dd0 +  1 + koff) * LP + hh) * LP + wc];
        b1.x = rot[((dd0 +  4 + koff) * LP + hh) * LP + wc];
        b1.y = rot[((dd0 +  5 + koff) * LP + hh) * LP + wc];
        b2.x = rot[((dd0 +  8 + koff) * LP + hh) * LP + wc];
        b2.y = rot[((dd0 +  9 + koff) * LP + hh) * LP + wc];
        b3.x = rot[((dd0 + 12 + koff) * LP + hh) * LP + wc];
        b3.y = rot[((dd0 + 13 + koff) * LP + hh) * LP + wc];

        v8f acc = {};
        acc = __builtin_amdgcn_wmma_f32_16x16x4_f32(false, a0, false, b0, (short)0, acc, false, false);
        acc = __builtin_amdgcn_wmma_f32_16x16x4_f32(false, a1, false, b1, (short)0, acc, false, false);
        acc = __builtin_amdgcn_wmma_f32_16x16x4_f32(false, a2, false, b2, (short)0, acc, false, false);
        acc = __builtin_amdgcn_wmma_f32_16x16x4_f32(false, a3, false, b3, (short)0, acc, false, false);

        // block total = Y[0][col], held in acc[0] of lane 'col' (lanes 0-15)
        float bsum = __shfl(acc[0], col, 32);

        #pragma unroll
        for (int v = 0; v < 8; ++v) {
            int k = v + 8 * half;                      // D layout: VGPR v -> row v / v+8
            float outv = __expf(-CABS * (acc[v] + carry));
            rot[((dd0 + k) * LP + hh) * LP + wc] = outv;
        }
        carry += bsum;
    }
}

// ---------------- Kernel C: back-rotation, central 129^3 only --------------
__global__ __launch_bounds__(256) void k_rotate_bwd(const float* __restrict__ T,
                                                    float* __restrict__ A, M3 R) {
    int idx = blockIdx.x * 256 + threadIdx.x;
    if (idx >= NS3) return;
    int xx = idx % S;
    int t  = idx / S;
    int yy = t % S;
    int zz = t / S;
    const float Gf   = 0.5f * (float)(LP - 1);
    const float invG = 1.0f / Gf;
    float bx = (xx + 48) * invG - 1.0f;
    float by = (yy + 48) * invG - 1.0f;
    float bz = (zz + 48) * invG - 1.0f;
    float xs = (R.m[0] * bx + R.m[1] * by + R.m[2] * bz + 1.0f) * Gf;
    float ys = (R.m[3] * bx + R.m[4] * by + R.m[5] * bz + 1.0f) * Gf;
    float zs = (R.m[6] * bx + R.m[7] * by + R.m[8] * bz + 1.0f) * Gf;
    float fx = floorf(xs), fy = floorf(ys), fz = floorf(zs);
    int x0 = (int)fx, y0 = (int)fy, z0 = (int)fz;
    float wx = xs - fx, wy = ys - fy, wz = zs - fz;
    float c000 = sample_L(T, z0,   y0,   x0  ), c001 = sample_L(T, z0,   y0,   x0+1);
    float c010 = sample_L(T, z0,   y0+1, x0  ), c011 = sample_L(T, z0,   y0+1, x0+1);
    float c100 = sample_L(T, z0+1, y0,   x0  ), c101 = sample_L(T, z0+1, y0,   x0+1);
    float c110 = sample_L(T, z0+1, y0+1, x0  ), c111 = sample_L(T, z0+1, y0+1, x0+1);
    float l00 = c000 + wx * (c001 - c000);
    float l01 = c010 + wx * (c011 - c010);
    float l10 = c100 + wx * (c101 - c100);
    float l11 = c110 + wx * (c111 - c110);
    float l0  = l00 + wy * (l01 - l00);
    float l1  = l10 + wy * (l11 - l10);
    A[idx] = l0 + wz * (l1 - l0);
}

// ---------------- Kernel D: fused view-transmittance + composite -----------
__global__ __launch_bounds__(256) void k_final(const float* __restrict__ d,
                                               const float* __restrict__ A0,
                                               const float* __restrict__ A1,
                                               float* __restrict__ out) {
    int idx = blockIdx.x * 256 + threadIdx.x;    // (h,w) column
    if (idx >= S * S) return;
    const float r0 = 1.0f,          g0 = 69.0f / 255.0f, b0 = 25.0f / 255.0f;
    const float r1 = 227.0f/255.0f, g1 = 1.0f,           b1 = 66.0f / 255.0f;
    float csum = 0.0f, aR = 0.0f, aG = 0.0f, aB = 0.0f;
    for (int z = S - 1; z >= 0; --z) {
        int o = z * S * S + idx;
        float dv = d[o];
        csum += dv;
        float tv = __expf(-CABS * csum);         // inclusive suffix transmittance
        float w  = dv * tv;
        float u0 = A0[o], u1 = A1[o];
        aR += w * (u0 * r0 + u1 * r1);
        aG += w * (u0 * g0 + u1 * g1);
        aB += w * (u0 * b0 + u1 * b1);
    }
    out[0 * S * S + idx] = fminf(fmaxf(CABS * aR, 0.0f), 1.0f);
    out[1 * S * S + idx] = fminf(fmaxf(CABS * aG, 0.0f), 1.0f);
    out[2 * S * S + idx] = fminf(fmaxf(CABS * aB, 0.0f), 1.0f);
}

// ---------------- host: rotation matrices per light ------------------------
static void make_rots(const double dir[3], M3* R1, M3* R2) {
    double n  = sqrt(dir[0]*dir[0] + dir[1]*dir[1] + dir[2]*dir[2]);
    double l0 = dir[0] / n, l1 = dir[1] / n, l2 = dir[2] / n;
    double y = -asin(l0);
    if (l2 < 0.0) y = -M_PI - y;
    double p = asin(l1);
    double cy = cos(y), sy = sin(y), cp = cos(p), sp = sin(p);
    // R1 = Ry(y) @ Rp(p)   (order 'ypr'); R2 = Rp(-p) @ Ry(-y) = R1^T
    float m[9] = {
        (float)cy,  (float)(sy*sp), (float)(sy*cp),
        0.0f,       (float)cp,      (float)(-sp),
        (float)(-sy),(float)(cy*sp),(float)(cy*cp)
    };
    for (int i = 0; i < 9; ++i) R1->m[i] = m[i];
    for (int r = 0; r < 3; ++r)
        for (int c = 0; c < 3; ++c)
            R2->m[r*3 + c] = m[c*3 + r];
}

extern "C" void kernel_launch(void* const* d_in, const int* in_sizes, int n_in,
                              void* d_out, int out_size, void* d_ws, size_t ws_size,
                              hipStream_t stream) {
    (void)in_sizes; (void)n_in; (void)out_size; (void)ws_size;
    const float* dvol = (const float*)d_in[0];
    float* out = (float*)d_out;

    char* ws = (char*)d_ws;
    size_t rotBytes = (size_t)NL3 * sizeof(float);       // ~45 MB, 256B aligned
    float* rot = (float*)ws;
    float* A0  = (float*)(ws + rotBytes);
    float* A1  = A0 + (size_t)NS3;

    const double dirs[2][3] = { {2.0, 1.0, 1.0}, {-1.0, 0.5, 0.0} };
    for (int l = 0; l < 2; ++l) {
        M3 R1, R2;
        make_rots(dirs[l], &R1, &R2);
        k_rotate_fwd<<<NL3 / 256, 256, 0, stream>>>(dvol, rot, R1);
        k_scan_exp<<<(LP * 14) / 8, 256, 0, stream>>>(rot);   // 3136 waves exactly
        float* A = l ? A1 : A0;
        k_rotate_bwd<<<(NS3 + 255) / 256, 256, 0, stream>>>(rot, A, R2);
    }
    k_final<<<(S * S + 255) / 256, 256, 0, stream>>>(dvol, A0, A1, out);
}